// DeepGraphInfomax_45208825757798
// MI455X (gfx1250) — compile-verified
//
#include <hip/hip_runtime.h>
#include <hip/hip_bf16.h>

typedef float v2f __attribute__((ext_vector_type(2)));
typedef float v8f __attribute__((ext_vector_type(8)));

#define N_ROWS 1000000
#define H_DIM  128
#define K_COMM 64
#define CLUSTER_TEMP 30.0f

// ---------------------------------------------------------------- zero ws
__global__ void dgi_zero_ws(float* ws, int n) {
  int i = blockIdx.x * blockDim.x + threadIdx.x;
  if (i < n) ws[i] = 0.0f;
}

// ------------------------------------------- pass 1: normalize + segment sum
// one wave32 per row: lane holds float4 (32*4 = 128 = H). LDS per-community
// accumulators (64x128 f32 = 32KB, fits the 320KB WGP LDS), ds_add_f32 atomics,
// single global atomic flush per block.
__global__ __launch_bounds__(256) void dgi_norm_accum(
    const float* __restrict__ z, const int* __restrict__ comm,
    float* __restrict__ zn, float* __restrict__ gsums, float* __restrict__ gcnt)
{
  __shared__ float lsum[K_COMM * H_DIM];
  __shared__ float lcnt[K_COMM];
  for (int i = threadIdx.x; i < K_COMM * H_DIM; i += 256) lsum[i] = 0.0f;
  if (threadIdx.x < K_COMM) lcnt[threadIdx.x] = 0.0f;
  __syncthreads();

  const int lane = threadIdx.x & 31;
  const int wave = threadIdx.x >> 5;
  const int gw   = blockIdx.x * 8 + wave;
  const int nw   = gridDim.x * 8;

  for (int row = gw; row < N_ROWS; row += nw) {
    const float4 v = *(const float4*)(z + (size_t)row * H_DIM + lane * 4);
    float ss = v.x * v.x + v.y * v.y + v.z * v.z + v.w * v.w;
    #pragma unroll
    for (int m = 16; m >= 1; m >>= 1) ss += __shfl_xor(ss, m, 32);
    const float rn = rsqrtf(ss);
    float4 o; o.x = v.x * rn; o.y = v.y * rn; o.z = v.z * rn; o.w = v.w * rn;
    *(float4*)(zn + (size_t)row * H_DIM + lane * 4) = o;

    const int c = comm[row];
    float* b = &lsum[c * H_DIM + lane * 4];
    atomicAdd(b + 0, o.x); atomicAdd(b + 1, o.y);
    atomicAdd(b + 2, o.z); atomicAdd(b + 3, o.w);
    if (lane == 0) atomicAdd(&lcnt[c], 1.0f);
  }
  __syncthreads();
  for (int i = threadIdx.x; i < K_COMM * H_DIM; i += 256) atomicAdd(&gsums[i], lsum[i]);
  if (threadIdx.x < K_COMM) atomicAdd(&gcnt[threadIdx.x], lcnt[threadIdx.x]);
}

// ---------------------------------------------------------- pass 2: mu = s/c
__global__ void dgi_mu(const float* __restrict__ gsums,
                       const float* __restrict__ gcnt,
                       float* __restrict__ mu) {
  int i = blockIdx.x * blockDim.x + threadIdx.x;
  if (i < K_COMM * H_DIM) mu[i] = gsums[i] / gcnt[i >> 7];
}

// ----------------------- pass 3: dist = zn @ mu^T (fp32 WMMA) + fused softmax
// One wave per 16-row tile; 4 D-tiles (16x16) cover all 64 communities.
// A frag (16x4 f32): lanes 0-15 -> row M=lane,   K = k0+0..1
//                    lanes16-31 -> row M=lane-16,K = k0+2..3   (float2/lane)
// B frag (4x16 f32): same pattern with N instead of M, sourced from mu in LDS.
// D tile: VGPR j: lanes0-15 -> (M=j, N=lane), lanes16-31 -> (M=j+8, N=lane-16)
// so per-row softmax = reduction over 16-lane halves + the 4 tiles.
__global__ __launch_bounds__(256) void dgi_dist_softmax(
    const float* __restrict__ zn, const float* __restrict__ mu,
    float* __restrict__ r_out, float* __restrict__ dist_out, int ntiles)
{
  __shared__ float muS[K_COMM * H_DIM];
  for (int i = threadIdx.x; i < K_COMM * H_DIM; i += 256) muS[i] = mu[i];
  __syncthreads();

  const int lane = threadIdx.x & 31;
  const int wave = threadIdx.x >> 5;
  const int tile = blockIdx.x * 8 + wave;
  if (tile >= ntiles) return;               // wave-uniform: EXEC stays all-1s

  const int rowbase = tile * 16;
  const int l15  = lane & 15;
  const int half = lane >> 4;
  const int koff = half * 2;

  v8f acc0 = {}, acc1 = {}, acc2 = {}, acc3 = {};

  const float* arow = zn + (size_t)(rowbase + l15) * H_DIM + koff;
  const float* b0 = &muS[( 0 + l15) * H_DIM + koff];
  const float* b1 = &muS[(16 + l15) * H_DIM + koff];
  const float* b2 = &muS[(32 + l15) * H_DIM + koff];
  const float* b3 = &muS[(48 + l15) * H_DIM + koff];

  #pragma unroll 4
  for (int k0 = 0; k0 < H_DIM; k0 += 4) {
    const v2f a  = *(const v2f*)(arow + k0);
    const v2f bb0 = *(const v2f*)(b0 + k0);
    const v2f bb1 = *(const v2f*)(b1 + k0);
    const v2f bb2 = *(const v2f*)(b2 + k0);
    const v2f bb3 = *(const v2f*)(b3 + k0);
    acc0 = __builtin_amdgcn_wmma_f32_16x16x4_f32(false, a, false, bb0, (short)0, acc0, false, false);
    acc1 = __builtin_amdgcn_wmma_f32_16x16x4_f32(false, a, false, bb1, (short)0, acc1, false, false);
    acc2 = __builtin_amdgcn_wmma_f32_16x16x4_f32(false, a, false, bb2, (short)0, acc2, false, false);
    acc3 = __builtin_amdgcn_wmma_f32_16x16x4_f32(false, a, false, bb3, (short)0, acc3, false, false);
  }

  // fused softmax epilogue: each (j, lane-half) pair is one row of dist
  #pragma unroll
  for (int j = 0; j < 8; ++j) {
    const int m = rowbase + j + half * 8;
    const float d0 = acc0[j], d1 = acc1[j], d2 = acc2[j], d3 = acc3[j];
    float v0 = CLUSTER_TEMP * d0, v1 = CLUSTER_TEMP * d1;
    float v2 = CLUSTER_TEMP * d2, v3 = CLUSTER_TEMP * d3;

    float mx = fmaxf(fmaxf(v0, v1), fmaxf(v2, v3));
    #pragma unroll
    for (int s = 8; s >= 1; s >>= 1) mx = fmaxf(mx, __shfl_xor(mx, s, 32));

    const float e0 = __expf(v0 - mx), e1 = __expf(v1 - mx);
    const float e2 = __expf(v2 - mx), e3 = __expf(v3 - mx);
    float sum = e0 + e1 + e2 + e3;
    #pragma unroll
    for (int s = 8; s >= 1; s >>= 1) sum += __shfl_xor(sum, s, 32);
    const float inv = 1.0f / sum;

    const size_t ob = (size_t)m * K_COMM + l15;
    dist_out[ob +  0] = d0; dist_out[ob + 16] = d1;
    dist_out[ob + 32] = d2; dist_out[ob + 48] = d3;
    r_out[ob +  0] = e0 * inv; r_out[ob + 16] = e1 * inv;
    r_out[ob + 32] = e2 * inv; r_out[ob + 48] = e3 * inv;
  }
}

// ------------------------------------------------------------------- launch
extern "C" void kernel_launch(void* const* d_in, const int* in_sizes, int n_in,
                              void* d_out, int out_size, void* d_ws, size_t ws_size,
                              hipStream_t stream) {
  const float* z    = (const float*)d_in[0];
  const int*   comm = (const int*)d_in[1];

  float* out  = (float*)d_out;
  float* zn   = out;                                   // [N, H]
  float* mu   = zn + (size_t)N_ROWS * H_DIM;           // [K, H]
  float* r    = mu + (size_t)K_COMM * H_DIM;           // [N, K]
  float* dist = r  + (size_t)N_ROWS * K_COMM;          // [N, K]

  float* gsums = (float*)d_ws;                         // [K, H]
  float* gcnt  = gsums + K_COMM * H_DIM;               // [K]

  const int nws = K_COMM * H_DIM + K_COMM;
  dgi_zero_ws<<<(nws + 255) / 256, 256, 0, stream>>>(gsums, nws);

  dgi_norm_accum<<<1024, 256, 0, stream>>>(z, comm, zn, gsums, gcnt);

  dgi_mu<<<(K_COMM * H_DIM + 255) / 256, 256, 0, stream>>>(gsums, gcnt, mu);

  const int ntiles = N_ROWS / 16;  // 62500, exact
  dgi_dist_softmax<<<(ntiles + 7) / 8, 256, 0, stream>>>(zn, mu, r, dist, ntiles);
}